// GCN_57440892616776
// MI455X (gfx1250) — compile-verified
//
#include <hip/hip_runtime.h>
#include <hip/hip_bf16.h>
#include <math.h>

typedef __attribute__((ext_vector_type(2))) float v2f;
typedef __attribute__((ext_vector_type(8))) float v8f;

#define N_NODES 100000
#define N_EDGES 3200000
#define F_IN    512
#define F_HID   16
#define F_OUT   40
#define N_TILES (N_NODES / 16)   // 6250 exactly

// ---------------------------------------------------------------- degree/norm
__global__ void k_fill_deg(float* __restrict__ deg) {
    int i = blockIdx.x * blockDim.x + threadIdx.x;
    if (i < N_NODES) deg[i] = 1.0f;               // self loop
}

__global__ void k_count(const long long* __restrict__ ei, float* __restrict__ deg) {
    int e = blockIdx.x * blockDim.x + threadIdx.x;
    if (e < N_EDGES) {
        int r = (int)ei[e];                       // edge_index[0][e]
        atomicAdd(deg + r, 1.0f);
    }
}

__global__ void k_dinv(float* __restrict__ deg) {
    int i = blockIdx.x * blockDim.x + threadIdx.x;
    if (i < N_NODES) deg[i] = rsqrtf(deg[i]);     // deg >= 1 always
}

// ---------------------------------------------------------------- GEMM1: h0 = x @ W1
// Wave-per-16-row-tile, V_WMMA_F32_16X16X4_F32, K=512 -> 128 WMMA per tile.
__global__ void __launch_bounds__(256)
k_gemm1(const float* __restrict__ x, const float* __restrict__ W1,
        float* __restrict__ h0) {
    __shared__ float w1s[F_IN * F_HID];           // 32 KB
    for (int i = threadIdx.x; i < F_IN * F_HID; i += blockDim.x) w1s[i] = W1[i];
    __syncthreads();

    const int wave = threadIdx.x >> 5;
    const int lane = threadIdx.x & 31;
    const int tile = blockIdx.x * (blockDim.x >> 5) + wave;
    if (tile >= N_TILES) return;

    const int m     = lane & 15;                  // output col / A row
    const int khalf = (lane >> 4) << 1;           // 0 or 2: K sub-pair per lane group
    const float* xrow = x + (size_t)(tile * 16 + m) * F_IN + khalf;

    v8f c = {0.f, 0.f, 0.f, 0.f, 0.f, 0.f, 0.f, 0.f};
    #pragma unroll 8
    for (int k0 = 0; k0 < F_IN; k0 += 4) {
        v2f a = *(const v2f*)(xrow + k0);         // (K=k0+khalf, +1) for row m
        const int kb = k0 + khalf;
        v2f b;
        b.x = w1s[kb * F_HID + m];                // B[kb][m]
        b.y = w1s[(kb + 1) * F_HID + m];          // B[kb+1][m]
        c = __builtin_amdgcn_wmma_f32_16x16x4_f32(false, a, false, b,
                                                  (short)0, c, false, false);
    }
    // C: VGPR v -> row = v + 8*(lane>=16), col = m
    const int rbase = tile * 16 + ((lane >> 4) << 3);
    float* out = h0 + (size_t)rbase * F_HID + m;
    #pragma unroll
    for (int v = 0; v < 8; ++v) out[v * F_HID] = c[v];
}

// ---------------------------------------------------------------- layer-1 init: bias + self loop
__global__ void k_init1(const float* __restrict__ h0, const float* __restrict__ b1,
                        const float* __restrict__ dinv, float* __restrict__ h1) {
    int t = blockIdx.x * blockDim.x + threadIdx.x;
    if (t >= N_NODES * F_HID) return;
    int i = t / F_HID, f = t - i * F_HID;
    float d = dinv[i];
    h1[t] = b1[f] + d * d * h0[t];
}

// ---------------------------------------------------------------- SpMM 16-wide
__global__ void k_spmm16(const long long* __restrict__ ei, const float* __restrict__ dinv,
                         const float* __restrict__ h0, float* __restrict__ h1) {
    int e = blockIdx.x * blockDim.x + threadIdx.x;
    if (e >= N_EDGES) return;
    int r = (int)ei[e];
    int c = (int)ei[N_EDGES + e];
    float w = dinv[r] * dinv[c];
    const float4* src = (const float4*)(h0 + (size_t)c * F_HID);
    float* dst = h1 + (size_t)r * F_HID;
    #pragma unroll
    for (int q = 0; q < 4; ++q) {
        float4 v = src[q];                         // L2-resident gather (h0 = 6.4 MB)
        atomicAdd(dst + q * 4 + 0, w * v.x);
        atomicAdd(dst + q * 4 + 1, w * v.y);
        atomicAdd(dst + q * 4 + 2, w * v.z);
        atomicAdd(dst + q * 4 + 3, w * v.w);
    }
}

// ---------------------------------------------------------------- GEMM2: g2 = relu(h1) @ W2
// K=16 (4 WMMA steps), N=40 padded to 48 (3 column tiles). ReLU fused into A load.
__global__ void __launch_bounds__(256)
k_gemm2(const float* __restrict__ h1, const float* __restrict__ W2,
        float* __restrict__ g2) {
    __shared__ float w2s[F_HID * 48];              // 3 KB, zero padded cols 40..47
    for (int i = threadIdx.x; i < F_HID * 48; i += blockDim.x) {
        int k = i / 48, n = i - k * 48;
        w2s[i] = (n < F_OUT) ? W2[k * F_OUT + n] : 0.0f;
    }
    __syncthreads();

    const int wave = threadIdx.x >> 5;
    const int lane = threadIdx.x & 31;
    const int tile = blockIdx.x * (blockDim.x >> 5) + wave;
    if (tile >= N_TILES) return;

    const int m     = lane & 15;
    const int khalf = (lane >> 4) << 1;
    const float* arow = h1 + (size_t)(tile * 16 + m) * F_HID + khalf;

    v2f a[4];
    #pragma unroll
    for (int s = 0; s < 4; ++s) {                  // preload & ReLU the A fragments once
        v2f av = *(const v2f*)(arow + s * 4);
        a[s].x = fmaxf(av.x, 0.0f);
        a[s].y = fmaxf(av.y, 0.0f);
    }

    const int rbase = tile * 16 + ((lane >> 4) << 3);
    #pragma unroll
    for (int nt = 0; nt < 3; ++nt) {
        v8f c = {0.f, 0.f, 0.f, 0.f, 0.f, 0.f, 0.f, 0.f};
        #pragma unroll
        for (int s = 0; s < 4; ++s) {
            const int kb = s * 4 + khalf;
            v2f b;
            b.x = w2s[kb * 48 + nt * 16 + m];
            b.y = w2s[(kb + 1) * 48 + nt * 16 + m];
            c = __builtin_amdgcn_wmma_f32_16x16x4_f32(false, a[s], false, b,
                                                      (short)0, c, false, false);
        }
        const int col = nt * 16 + m;
        if (col < F_OUT) {                         // guard store only; WMMA ran full-EXEC
            float* out = g2 + (size_t)rbase * F_OUT + col;
            #pragma unroll
            for (int v = 0; v < 8; ++v) out[v * F_OUT] = c[v];
        }
    }
}

// ---------------------------------------------------------------- layer-2 init: bias + self loop
__global__ void k_init2(const float* __restrict__ g2, const float* __restrict__ b2,
                        const float* __restrict__ dinv, float* __restrict__ out) {
    int t = blockIdx.x * blockDim.x + threadIdx.x;
    if (t >= N_NODES * F_OUT) return;
    int i = t / F_OUT, f = t - i * F_OUT;
    float d = dinv[i];
    out[t] = b2[f] + d * d * g2[t];
}

// ---------------------------------------------------------------- SpMM 40-wide
__global__ void k_spmm40(const long long* __restrict__ ei, const float* __restrict__ dinv,
                         const float* __restrict__ g2, float* __restrict__ out) {
    int e = blockIdx.x * blockDim.x + threadIdx.x;
    if (e >= N_EDGES) return;
    int r = (int)ei[e];
    int c = (int)ei[N_EDGES + e];
    float w = dinv[r] * dinv[c];
    const float4* src = (const float4*)(g2 + (size_t)c * F_OUT);  // 160B rows, 16B aligned
    float* dst = out + (size_t)r * F_OUT;
    #pragma unroll
    for (int q = 0; q < 10; ++q) {
        float4 v = src[q];                          // L2-resident gather (g2 = 16 MB)
        atomicAdd(dst + q * 4 + 0, w * v.x);
        atomicAdd(dst + q * 4 + 1, w * v.y);
        atomicAdd(dst + q * 4 + 2, w * v.z);
        atomicAdd(dst + q * 4 + 3, w * v.w);
    }
}

// ---------------------------------------------------------------- log_softmax, wave32 per row
__global__ void __launch_bounds__(256)
k_logsoftmax(float* __restrict__ out) {
    const int wave = threadIdx.x >> 5;
    const int lane = threadIdx.x & 31;
    const int row  = blockIdx.x * (blockDim.x >> 5) + wave;
    if (row >= N_NODES) return;
    float* p = out + (size_t)row * F_OUT;

    float v0 = p[lane];
    float v1 = (lane < 8) ? p[32 + lane] : -INFINITY;

    float mx = fmaxf(v0, v1);
    #pragma unroll
    for (int off = 16; off > 0; off >>= 1)
        mx = fmaxf(mx, __shfl_xor(mx, off, 32));

    float s = __expf(v0 - mx) + ((lane < 8) ? __expf(v1 - mx) : 0.0f);
    #pragma unroll
    for (int off = 16; off > 0; off >>= 1)
        s += __shfl_xor(s, off, 32);

    float lg = mx + __logf(s);
    p[lane] = v0 - lg;
    if (lane < 8) p[32 + lane] = v1 - lg;
}

// ---------------------------------------------------------------- host launcher
extern "C" void kernel_launch(void* const* d_in, const int* in_sizes, int n_in,
                              void* d_out, int out_size, void* d_ws, size_t ws_size,
                              hipStream_t stream) {
    const float*     x  = (const float*)d_in[0];
    const float*     W1 = (const float*)d_in[1];
    const float*     b1 = (const float*)d_in[2];
    const float*     W2 = (const float*)d_in[3];
    const float*     b2 = (const float*)d_in[4];
    const long long* ei = (const long long*)d_in[5];
    float* out = (float*)d_out;

    // workspace layout (float elements, 16-float aligned chunks): ~29.2 MB total
    float* wsf  = (float*)d_ws;
    float* dinv = wsf;                    // 100000
    float* h0   = wsf + 100352;           // 100000*16
    float* h1   = h0  + 1600000;
    float* g2   = h1  + 1600000;          // 100000*40

    const int B = 256;
    const int gNodes  = (N_NODES + B - 1) / B;            // 391
    const int gEdges  = (N_EDGES + B - 1) / B;            // 12500
    const int gTiles  = (N_TILES + 7) / 8;                // 782 (8 waves/block)
    const int gH1     = (N_NODES * F_HID + B - 1) / B;    // 6250
    const int gOut    = (N_NODES * F_OUT + B - 1) / B;    // 15625
    const int gRows   = (N_NODES + 7) / 8;                // 12500 (wave per row)

    k_fill_deg  <<<gNodes, B, 0, stream>>>(dinv);
    k_count     <<<gEdges, B, 0, stream>>>(ei, dinv);
    k_dinv      <<<gNodes, B, 0, stream>>>(dinv);

    k_gemm1     <<<gTiles, B, 0, stream>>>(x, W1, h0);
    k_init1     <<<gH1,    B, 0, stream>>>(h0, b1, dinv, h1);
    k_spmm16    <<<gEdges, B, 0, stream>>>(ei, dinv, h0, h1);

    k_gemm2     <<<gTiles, B, 0, stream>>>(h1, W2, g2);
    k_init2     <<<gOut,   B, 0, stream>>>(g2, b2, dinv, out);
    k_spmm40    <<<gEdges, B, 0, stream>>>(ei, dinv, g2, out);

    k_logsoftmax<<<gRows,  B, 0, stream>>>(out);
}